// MyoNet_79087527789010
// MI455X (gfx1250) — compile-verified
//
#include <hip/hip_runtime.h>
#include <hip/hip_bf16.h>
#include <math.h>

#define F_IN  80
#define F_OUT 40
#define HID   20
#define NC    5

typedef __attribute__((ext_vector_type(2))) float v2f;
typedef __attribute__((ext_vector_type(8))) float v8f;

// ---------------------------------------------------------------- zero fill
__global__ void zero_kernel(float* p, long long n) {
    long long i = (long long)blockIdx.x * blockDim.x + threadIdx.x;
    if (i < n) p[i] = 0.0f;
}

// ---------------------------------------------------------- deg[src] += w
__global__ void deg_kernel(const long long* ei, const float* ew, float* deg, int E) {
    int e = blockIdx.x * blockDim.x + threadIdx.x;
    if (e < E) atomicAdd(&deg[(int)ei[e]], ew[e]);
}

// ------------------------------------------- deg -> dinv = rsqrt(deg) in place
__global__ void dinv_kernel(float* deg, int N) {
    int i = blockIdx.x * blockDim.x + threadIdx.x;
    if (i < N) {
        float d = deg[i];
        deg[i] = d > 0.0f ? rsqrtf(d) : 0.0f;
    }
}

// ---------------- Tx1[dst] += (-dinv[src]*w*dinv[dst]) * x[src]  (80 feats)
// 20 threads per edge, each handles 4 consecutive features (float4 gather).
__global__ void scatter_kernel(const long long* ei, const float* ew, const float* dinv,
                               const float* x, float* Tx1, int E) {
    long long tid = (long long)blockIdx.x * blockDim.x + threadIdx.x;
    long long e = tid / 20;
    int c = (int)(tid % 20);
    if (e >= E) return;
    int s = (int)ei[e];
    int d = (int)ei[(long long)E + e];
    float coef = -dinv[s] * ew[e] * dinv[d];
    const float4 xv = *(const float4*)(x + (long long)s * F_IN + c * 4);
    float* tp = Tx1 + (long long)d * F_IN + c * 4;
    atomicAdd(tp + 0, coef * xv.x);
    atomicAdd(tp + 1, coef * xv.y);
    atomicAdd(tp + 2, coef * xv.z);
    atomicAdd(tp + 3, coef * xv.w);
}

// ---------------- H = relu(X*W0 + Tx1*W1 + b)  via V_WMMA_F32_16X16X4_F32
// One wave computes one 16x16 output tile. 3 column tiles cover F_OUT=40 (padded).
__global__ void gemm_relu_wmma(const float* __restrict__ X, const float* __restrict__ T1,
                               const float* __restrict__ W0, const float* __restrict__ W1,
                               const float* __restrict__ bias,
                               float* __restrict__ Hout, int N, int MT) {
    int wid  = (blockIdx.x * blockDim.x + threadIdx.x) >> 5;   // wave id (uniform)
    int lane = threadIdx.x & 31;
    int mt = wid / 3;
    int nt = wid - mt * 3;
    if (mt >= MT) return;                                      // wave-uniform exit

    int half = lane >> 4;       // 0: lanes 0-15, 1: lanes 16-31
    int l    = lane & 15;
    int row  = mt * 16 + l;                                    // A row for this lane
    int arow = row < N ? row : N - 1;                          // clamp (no divergence)
    int col  = nt * 16 + l;                                    // B/C column for this lane
    int ccol = col < F_OUT ? col : 0;
    float cmask = col < F_OUT ? 1.0f : 0.0f;

#if defined(__HIP_DEVICE_COMPILE__) && __has_builtin(__builtin_amdgcn_wmma_f32_16x16x4_f32)
    v8f acc = {};
    for (int pass = 0; pass < 2; ++pass) {
        const float* Ap = pass ? T1 : X;
        const float* Wp = pass ? W1 : W0;
        #pragma unroll 4
        for (int k0 = 0; k0 < F_IN; k0 += 4) {
            int ka = k0 + 2 * half;   // K indices handled by this lane-half
            v2f a, b;
            a.x = Ap[(long long)arow * F_IN + ka];
            a.y = Ap[(long long)arow * F_IN + ka + 1];
            b.x = Wp[ka * F_OUT + ccol] * cmask;
            b.y = Wp[(ka + 1) * F_OUT + ccol] * cmask;
            acc = __builtin_amdgcn_wmma_f32_16x16x4_f32(
                /*neg_a=*/false, a, /*neg_b=*/false, b,
                /*c_mod=*/(short)0, acc, /*reuse_a=*/false, /*reuse_b=*/false);
        }
    }
    if (col < F_OUT) {
        float bv = bias[col];
        #pragma unroll
        for (int r = 0; r < 8; ++r) {
            int ro = mt * 16 + r + 8 * half;    // C/D layout: VGPR r -> M = r + 8*half
            if (ro < N) {
                float v = acc[r] + bv;
                Hout[(long long)ro * F_OUT + col] = v > 0.0f ? v : 0.0f;
            }
        }
    }
#else
    // Scalar fallback (host pass / missing builtin): same outputs per lane.
    float accs[8] = {0, 0, 0, 0, 0, 0, 0, 0};
    for (int pass = 0; pass < 2; ++pass) {
        const float* Ap = pass ? T1 : X;
        const float* Wp = pass ? W1 : W0;
        for (int r = 0; r < 8; ++r) {
            int ro = mt * 16 + r + 8 * half;
            int rr = ro < N ? ro : N - 1;
            float s = 0.0f;
            for (int k = 0; k < F_IN; ++k)
                s += Ap[(long long)rr * F_IN + k] * Wp[k * F_OUT + ccol];
            accs[r] += s;
        }
    }
    if (col < F_OUT) {
        float bv = bias[col];
        for (int r = 0; r < 8; ++r) {
            int ro = mt * 16 + r + 8 * half;
            if (ro < N) {
                float v = accs[r] + bv;
                Hout[(long long)ro * F_OUT + col] = v > 0.0f ? v : 0.0f;
            }
        }
    }
#endif
}

// ---------------------------------------- pooled[batch[i]] += H[i] (40 feats)
__global__ void pool_kernel(const float* __restrict__ Hin, const long long* batch,
                            float* pooled, int N) {
    long long tid = (long long)blockIdx.x * blockDim.x + threadIdx.x;
    long long i = tid / F_OUT;
    int f = (int)(tid % F_OUT);
    if (i >= N) return;
    atomicAdd(&pooled[(int)batch[i] * F_OUT + f], Hin[i * F_OUT + f]);
}

// ------------- sequential RNN(40->20,tanh) over G steps + FC(20->5) + logsoftmax
// Single wave32 block; weights cached in LDS.
__global__ void rnn_head_kernel(const float* __restrict__ pooled,
                                const float* W_ih, const float* W_hh,
                                const float* b_ih, const float* b_hh,
                                const float* W_fc, const float* b_fc,
                                float* __restrict__ out, int G) {
    __shared__ float sWih[HID * F_OUT];
    __shared__ float sWhh[HID * HID];
    __shared__ float sWfc[NC * HID];
    __shared__ float sbih[HID], sbhh[HID], sbfc[NC];
    __shared__ float hbuf[HID];
    __shared__ float xbuf[F_OUT];
    __shared__ float lbuf[NC];
    int t = threadIdx.x;
    for (int i = t; i < HID * F_OUT; i += 32) sWih[i] = W_ih[i];
    for (int i = t; i < HID * HID;  i += 32) sWhh[i] = W_hh[i];
    for (int i = t; i < NC * HID;   i += 32) sWfc[i] = W_fc[i];
    if (t < HID) { sbih[t] = b_ih[t]; sbhh[t] = b_hh[t]; hbuf[t] = 0.0f; }
    if (t < NC)  sbfc[t] = b_fc[t];
    __syncthreads();

    for (int g = 0; g < G; ++g) {
        for (int i = t; i < F_OUT; i += 32) xbuf[i] = pooled[(long long)g * F_OUT + i];
        __syncthreads();
        float hn = 0.0f;
        if (t < HID) {
            float s = sbih[t] + sbhh[t];
            #pragma unroll 8
            for (int k = 0; k < F_OUT; ++k) s += sWih[t * F_OUT + k] * xbuf[k];
            #pragma unroll 4
            for (int k = 0; k < HID; ++k)  s += sWhh[t * HID + k] * hbuf[k];
            hn = tanhf(s);
        }
        __syncthreads();                 // all hbuf reads done
        if (t < HID) hbuf[t] = hn;
        __syncthreads();
        if (t < NC) {
            float lg = sbfc[t];
            #pragma unroll
            for (int k = 0; k < HID; ++k) lg += sWfc[t * HID + k] * hbuf[k];
            lbuf[t] = lg;
        }
        __syncthreads();
        if (t < NC) {
            float m = lbuf[0];
            for (int c = 1; c < NC; ++c) m = fmaxf(m, lbuf[c]);
            float ssum = 0.0f;
            for (int c = 0; c < NC; ++c) ssum += expf(lbuf[c] - m);
            out[(long long)g * NC + t] = lbuf[t] - m - logf(ssum);
        }
        __syncthreads();
    }
}

extern "C" void kernel_launch(void* const* d_in, const int* in_sizes, int n_in,
                              void* d_out, int out_size, void* d_ws, size_t ws_size,
                              hipStream_t stream) {
    const float*      x     = (const float*)d_in[0];
    const long long*  ei    = (const long long*)d_in[1];
    const float*      ew    = (const float*)d_in[2];
    const long long*  batch = (const long long*)d_in[3];
    const float*      W0    = (const float*)d_in[5];
    const float*      W1    = (const float*)d_in[6];
    const float*      bconv = (const float*)d_in[7];
    const float*      Wih   = (const float*)d_in[8];
    const float*      Whh   = (const float*)d_in[9];
    const float*      bih   = (const float*)d_in[10];
    const float*      bhh   = (const float*)d_in[11];
    const float*      Wfc   = (const float*)d_in[12];
    const float*      bfc   = (const float*)d_in[13];

    int N = in_sizes[0] / F_IN;
    int E = in_sizes[2];
    int G = out_size / NC;

    float* ws     = (float*)d_ws;
    float* deg    = ws;                                   // N (becomes dinv in place)
    float* Tx1    = ws + N;                               // N*F_IN
    float* Hbuf   = Tx1 + (long long)N * F_IN;            // N*F_OUT
    float* pooled = Hbuf + (long long)N * F_OUT;          // G*F_OUT

    long long zc = (long long)N + (long long)N * F_IN;    // deg + Tx1 contiguous
    zero_kernel<<<(unsigned)((zc + 255) / 256), 256, 0, stream>>>(ws, zc);
    long long pc = (long long)G * F_OUT;
    zero_kernel<<<(unsigned)((pc + 255) / 256), 256, 0, stream>>>(pooled, pc);

    deg_kernel<<<(E + 255) / 256, 256, 0, stream>>>(ei, ew, deg, E);
    dinv_kernel<<<(N + 255) / 256, 256, 0, stream>>>(deg, N);

    long long sth = (long long)E * 20;
    scatter_kernel<<<(unsigned)((sth + 255) / 256), 256, 0, stream>>>(ei, ew, deg, x, Tx1, E);

    int MT = (N + 15) / 16;
    int tiles = MT * 3;                    // 3 column tiles cover 40 outputs
    int blocks = (tiles + 7) / 8;          // 8 waves per 256-thread block
    gemm_relu_wmma<<<blocks, 256, 0, stream>>>(x, Tx1, W0, W1, bconv, Hbuf, N, MT);

    long long pth = (long long)N * F_OUT;
    pool_kernel<<<(unsigned)((pth + 255) / 256), 256, 0, stream>>>(Hbuf, batch, pooled, N);

    rnn_head_kernel<<<1, 32, 0, stream>>>(pooled, Wih, Whh, bih, bhh, Wfc, bfc,
                                          (float*)d_out, G);
}